// Kernel_15812022909139887089_53472342835841
// MI455X (gfx1250) — compile-verified
//
#include <hip/hip_runtime.h>
#include <hip/hip_bf16.h>
#include <math.h>
#include <stdint.h>

// Problem constants (from reference): N=8, C=256, H=W=56
constexpr int kN  = 8;
constexpr int kC  = 256;
constexpr int kH  = 56;
constexpr int kW  = 56;
constexpr int kHW = kH * kW;      // 3136
constexpr int kC8 = kC / 8;       // 32
constexpr int kK9 = 9 * kC;       // 2304

typedef float v2f __attribute__((ext_vector_type(2)));
typedef float v8f __attribute__((ext_vector_type(8)));

// D = A(16x4) * B(4x16) + C  -- fp32 WMMA, wave32
#define WMMA_F32(acc, a, b)                                                    \
  acc = __builtin_amdgcn_wmma_f32_16x16x4_f32(false, (a), false, (b),          \
                                              (short)0, (acc), false, false)

// ---------------------------------------------------------------------------
// t3 = (p2 * x)^2   (p2 broadcast over batch)
// ---------------------------------------------------------------------------
__global__ void k_t3(const float* __restrict__ x, const float* __restrict__ p2,
                     float* __restrict__ t3, int total) {
  int i = blockIdx.x * blockDim.x + threadIdx.x;
  if (i >= total) return;
  int chw = i % (kC * kHW);
  float v = p2[chw] * x[i];
  t3[i] = v * v;
}

// ---------------------------------------------------------------------------
// t4 = roll(roll(t3,+1,H),-1,W);  t5 = softmax(t4, axis=H)
// ---------------------------------------------------------------------------
__global__ void k_softmax(const float* __restrict__ t3, float* __restrict__ t5) {
  int i = blockIdx.x * blockDim.x + threadIdx.x;
  if (i >= kN * kC * kW) return;
  int w = i % kW;
  int c = (i / kW) % kC;
  int n = i / (kW * kC);
  int wp = (w + 1) % kW;
  const float* base = t3 + ((size_t)n * kC + c) * kHW;
  float mx = -1e30f;
  for (int h = 0; h < kH; ++h) {
    int hm = (h == 0) ? (kH - 1) : (h - 1);
    mx = fmaxf(mx, base[hm * kW + wp]);
  }
  float s = 0.f;
  for (int h = 0; h < kH; ++h) {
    int hm = (h == 0) ? (kH - 1) : (h - 1);
    s += __expf(base[hm * kW + wp] - mx);
  }
  float inv = 1.f / s;
  float* out = t5 + ((size_t)n * kC + c) * kHW + w;
  for (int h = 0; h < kH; ++h) {
    int hm = (h == 0) ? (kH - 1) : (h - 1);
    out[h * kW] = __expf(base[hm * kW + wp] - mx) * inv;
  }
}

// ---------------------------------------------------------------------------
// t7[n,o,p] = sum_{c,i,j} w7[o, c*9+i*3+j] * x[n,c, h+3(i-1), w+3(j-1)]
// One 512-thread block (16 waves = 16 o-tiles) per 16-pixel tile.
// B slab [9 shifts][64 ch][16 px] staged in LDS via async global->LDS copies
// (ASYNCcnt) and shared by all 16 waves; zero padding pre-seeded in LDS.
// ---------------------------------------------------------------------------
#define T7_CHUNK 64
#define T7_SLAB  (9 * T7_CHUNK * 16)   // 9216 floats = 36 KB
__global__ __launch_bounds__(512) void k_t7(const float* __restrict__ x,
                                            const float* __restrict__ w7,
                                            float* __restrict__ t7) {
  __shared__ float shB[T7_SLAB];
  int tid    = threadIdx.x;
  int lane   = tid & 31;
  int waveid = tid >> 5;                    // 0..15 -> o-tile
  int m = lane & 15, half = lane >> 4, kb = half << 1;
  int ptile = blockIdx.x, n = blockIdx.z;
  const float* xn = x + (size_t)n * kC * kHW;

  // Pre-zero LDS: invalid (s,px) slots stay zero for every channel chunk.
  for (int e = tid; e < T7_SLAB; e += 512) shB[e] = 0.f;
  __syncthreads();

  const float* wA = w7 + (size_t)(waveid * 16 + m) * kK9;  // A row = out channel
  v8f acc = {};
  for (int c0 = 0; c0 < kC; c0 += T7_CHUNK) {
    // ---- stage x chunk into LDS (gfx1250 async global->LDS, EXEC-masked) ----
    for (int e = tid; e < T7_SLAB; e += 512) {
      int s   = e >> 10;                    // / (T7_CHUNK*16)
      int rem = e & 1023;
      int c   = rem >> 4;
      int px  = rem & 15;
      int pix = ptile * 16 + px;
      int h = pix / kW, w = pix % kW;
      int hh = h + 3 * (s / 3 - 1);
      int ww = w + 3 * (s % 3 - 1);
      bool valid = (hh >= 0) & (hh < kH) & (ww >= 0) & (ww < kW);
      if (valid) {
        unsigned lds_addr = (unsigned)(uintptr_t)(&shB[e]);
        unsigned long long gaddr =
            (unsigned long long)(uintptr_t)(xn + (size_t)(c0 + c) * kHW +
                                            hh * kW + ww);
        asm volatile("global_load_async_to_lds_b32 %0, %1, off"
                     :: "v"(lds_addr), "v"(gaddr) : "memory");
      }
    }
    asm volatile("s_wait_asynccnt 0x0" ::: "memory");
    __syncthreads();
    // ---- consume: 9 shifts x 16 K-steps of fp32 WMMA ----
    for (int s = 0; s < 9; ++s) {
      const float* wAs = wA + s;
      const float* shs = shB + s * (T7_CHUNK * 16);
      for (int k = 0; k < T7_CHUNK; k += 4) {
        v2f a, b;
        a.x = wAs[(c0 + k + kb) * 9];
        a.y = wAs[(c0 + k + kb + 1) * 9];
        b.x = shs[(k + kb) * 16 + m];
        b.y = shs[(k + kb + 1) * 16 + m];
        WMMA_F32(acc, a, b);
      }
    }
    __syncthreads();   // protect LDS before next chunk's copies
  }
  float* out = t7 + (size_t)n * kC * kHW + ptile * 16 + m;
  for (int r = 0; r < 8; ++r) {
    int row = waveid * 16 + r + half * 8;
    out[(size_t)row * kHW] = acc[r];
  }
}

// ---------------------------------------------------------------------------
// t15[n,o,p] = sum_c w15[o,c] * x[n,c,(h-1)%H,w];  t17 = t3 - t15
// ---------------------------------------------------------------------------
__global__ __launch_bounds__(32) void k_t17(const float* __restrict__ x,
                                            const float* __restrict__ w15,
                                            const float* __restrict__ t3,
                                            float* __restrict__ t17) {
  int lane = threadIdx.x;
  int m = lane & 15, half = lane >> 4, kb = half << 1;
  int ptile = blockIdx.x, otile = blockIdx.y, n = blockIdx.z;
  int pix = ptile * 16 + m;
  int h = pix / kW, w = pix % kW;
  int hm = (h == 0) ? (kH - 1) : (h - 1);
  int poff = hm * kW + w;
  const float* xn = x + (size_t)n * kC * kHW;
  const float* wA = w15 + (size_t)(otile * 16 + m) * kC;
  v8f acc = {};
  for (int c = 0; c < kC; c += 4) {
    v2f a = *(const v2f*)(wA + c + kb);           // contiguous in K -> b64
    v2f b;
    b.x = xn[(size_t)(c + kb) * kHW + poff];
    b.y = xn[(size_t)(c + kb + 1) * kHW + poff];
    WMMA_F32(acc, a, b);
  }
  const float* t3n = t3 + (size_t)n * kC * kHW + ptile * 16 + m;
  float* out = t17 + (size_t)n * kC * kHW + ptile * 16 + m;
  for (int r = 0; r < 8; ++r) {
    size_t row = (size_t)(otile * 16 + r + half * 8) * kHW;
    out[row] = t3n[row] - acc[r];
  }
}

// ---------------------------------------------------------------------------
// out = t18 = t7 * t17  (t20 accumulated on top later)
// ---------------------------------------------------------------------------
__global__ void k_t18(const float* __restrict__ t7, const float* __restrict__ t17,
                      float* __restrict__ out, int total) {
  int i = blockIdx.x * blockDim.x + threadIdx.x;
  if (i < total) out[i] = t7[i] * t17[i];
}

// ---------------------------------------------------------------------------
// t8[n,c,d] = (1/56) * sum_s t5[n,c,s] * t3[n,d,s]
// ---------------------------------------------------------------------------
__global__ __launch_bounds__(32) void k_t8(const float* __restrict__ t5,
                                           const float* __restrict__ t3,
                                           float* __restrict__ t8) {
  int lane = threadIdx.x;
  int m = lane & 15, half = lane >> 4, kb = half << 1;
  int dtile = blockIdx.x, ctile = blockIdx.y, n = blockIdx.z;
  const float* arow = t5 + ((size_t)n * kC + ctile * 16 + m) * kHW;
  const float* brow = t3 + ((size_t)n * kC + dtile * 16 + m) * kHW;
  v8f acc = {};
  for (int s = 0; s < kHW; s += 4) {
    __builtin_prefetch(arow + s + 256, 0, 3);     // global_prefetch_b8
    __builtin_prefetch(brow + s + 256, 0, 3);
    v2f a = *(const v2f*)(arow + s + kb);          // b64 loads
    v2f b = *(const v2f*)(brow + s + kb);
    WMMA_F32(acc, a, b);
  }
  const float s_hw = 1.f / 56.f;   // 1/sqrt(H*W)
  float* out = t8 + (size_t)n * kC * kC + dtile * 16 + m;
  for (int r = 0; r < 8; ++r) {
    int row = ctile * 16 + r + half * 8;
    out[(size_t)row * kC] = acc[r] * s_hw;
  }
}

// ---------------------------------------------------------------------------
// t16[n,d,c] = sum_b t8[n,b,c] * p16[b,d]      (d over HW, c over C)
// ---------------------------------------------------------------------------
__global__ __launch_bounds__(32) void k_t16(const float* __restrict__ t8,
                                            const float* __restrict__ p16,
                                            float* __restrict__ t16) {
  int lane = threadIdx.x;
  int m = lane & 15, half = lane >> 4, kb = half << 1;
  int dtile = blockIdx.x, ctile = blockIdx.y, n = blockIdx.z;
  const float* acol = p16 + dtile * 16 + m;                       // + b*kHW
  const float* bmat = t8 + (size_t)n * kC * kC + ctile * 16 + m;  // + b*kC
  v8f acc = {};
  for (int b0 = 0; b0 < kC; b0 += 4) {
    v2f a, b;
    a.x = acol[(size_t)(b0 + kb) * kHW];
    a.y = acol[(size_t)(b0 + kb + 1) * kHW];
    b.x = bmat[(size_t)(b0 + kb) * kC];
    b.y = bmat[(size_t)(b0 + kb + 1) * kC];
    WMMA_F32(acc, a, b);
  }
  float* out = t16 + (size_t)n * kHW * kC + ctile * 16 + m;
  for (int r = 0; r < 8; ++r) {
    int row = dtile * 16 + r + half * 8;
    out[(size_t)row * kC] = acc[r];
  }
}

// ---------------------------------------------------------------------------
// t19[n,c,d] = (1/56) * sum_s t17[n,c,s] * t16[n,s,d]
// ---------------------------------------------------------------------------
__global__ __launch_bounds__(32) void k_t19(const float* __restrict__ t17,
                                            const float* __restrict__ t16,
                                            float* __restrict__ t19) {
  int lane = threadIdx.x;
  int m = lane & 15, half = lane >> 4, kb = half << 1;
  int dtile = blockIdx.x, ctile = blockIdx.y, n = blockIdx.z;
  const float* arow = t17 + ((size_t)n * kC + ctile * 16 + m) * kHW;
  const float* bcol = t16 + (size_t)n * kHW * kC + dtile * 16 + m;  // + s*kC
  v8f acc = {};
  for (int s = 0; s < kHW; s += 4) {
    __builtin_prefetch(arow + s + 256, 0, 3);
    v2f a = *(const v2f*)(arow + s + kb);          // contiguous in K
    v2f b;
    b.x = bcol[(size_t)(s + kb) * kC];
    b.y = bcol[(size_t)(s + kb + 1) * kC];
    WMMA_F32(acc, a, b);
  }
  const float s_hw = 1.f / 56.f;
  float* out = t19 + (size_t)n * kC * kC + dtile * 16 + m;
  for (int r = 0; r < 8; ++r) {
    int row = ctile * 16 + r + half * 8;
    out[(size_t)row * kC] = acc[r] * s_hw;
  }
}

// ---------------------------------------------------------------------------
// t6 -> t9 -> t11 fused: t11[n,p] = sum_b p11[b]*p9[b,p]*(sum_c w6[b,c]*t3[n,c,p])
// ---------------------------------------------------------------------------
__global__ void k_t11(const float* __restrict__ t3, const float* __restrict__ w6,
                      const float* __restrict__ p9, const float* __restrict__ p11,
                      float* __restrict__ t11) {
  int i = blockIdx.x * blockDim.x + threadIdx.x;
  if (i >= kN * kHW) return;
  int p = i % kHW, n = i / kHW;
  const float* t3n = t3 + (size_t)n * kC * kHW + p;
  float acc[kC8];
  for (int b = 0; b < kC8; ++b) acc[b] = 0.f;
  for (int c = 0; c < kC; ++c) {
    float v = t3n[(size_t)c * kHW];
#pragma unroll
    for (int b = 0; b < kC8; ++b) acc[b] = fmaf(w6[b * kC + c], v, acc[b]);
  }
  float s = 0.f;
  for (int b = 0; b < kC8; ++b) s += p11[b] * p9[(size_t)b * kHW + p] * acc[b];
  t11[i] = s;
}

// ---------------------------------------------------------------------------
// t10 = max(t5,t7); t12 = depthwise (3,1) dilation (2,1) pad (2,0) over H
// t13 = t11 - t12
// ---------------------------------------------------------------------------
__global__ void k_t13(const float* __restrict__ t5, const float* __restrict__ t7,
                      const float* __restrict__ w12, const float* __restrict__ t11,
                      float* __restrict__ t13, int total) {
  int i = blockIdx.x * blockDim.x + threadIdx.x;
  if (i >= total) return;
  int w = i % kW;
  int h = (i / kW) % kH;
  int c = (i / kHW) % kC;
  int n = i / (kC * kHW);
  float s = 0.f;
  for (int r = 0; r < 3; ++r) {
    int hh = h + 2 * (r - 1);
    if (hh >= 0 && hh < kH) {
      size_t idx = ((size_t)(n * kC + c)) * kHW + hh * kW + w;
      float v = fmaxf(t5[idx], t7[idx]);
      s = fmaf(w12[c * 3 + r], v, s);
    }
  }
  t13[i] = t11[(size_t)n * kHW + h * kW + w] - s;
}

// ---------------------------------------------------------------------------
// t20[n,c,s] = (1/16) * sum_d t19[n,d,c] * t13[n,d,s];  out += t20
// ---------------------------------------------------------------------------
__global__ __launch_bounds__(32) void k_t20(const float* __restrict__ t19,
                                            const float* __restrict__ t13,
                                            float* __restrict__ out) {
  int lane = threadIdx.x;
  int m = lane & 15, half = lane >> 4, kb = half << 1;
  int stile = blockIdx.x, ctile = blockIdx.y, n = blockIdx.z;
  const float* acol = t19 + (size_t)n * kC * kC + ctile * 16 + m;   // + d*kC
  const float* bcol = t13 + (size_t)n * kC * kHW + stile * 16 + m;  // + d*kHW
  v8f acc = {};
  for (int d = 0; d < kC; d += 4) {
    v2f a, b;
    a.x = acol[(size_t)(d + kb) * kC];
    a.y = acol[(size_t)(d + kb + 1) * kC];
    b.x = bcol[(size_t)(d + kb) * kHW];
    b.y = bcol[(size_t)(d + kb + 1) * kHW];
    WMMA_F32(acc, a, b);
  }
  const float s_c = 1.f / 16.f;   // 1/sqrt(C)
  float* o = out + (size_t)n * kC * kHW + stile * 16 + m;
  for (int r = 0; r < 8; ++r) {
    size_t row = (size_t)(ctile * 16 + r + half * 8) * kHW;
    o[row] += acc[r] * s_c;
  }
}

// ---------------------------------------------------------------------------
extern "C" void kernel_launch(void* const* d_in, const int* in_sizes, int n_in,
                              void* d_out, int out_size, void* d_ws, size_t ws_size,
                              hipStream_t stream) {
  (void)in_sizes; (void)n_in; (void)out_size; (void)ws_size;
  const float* x   = (const float*)d_in[0];
  const float* p2  = (const float*)d_in[1];
  const float* w6  = (const float*)d_in[2];
  const float* w7  = (const float*)d_in[3];
  const float* p9  = (const float*)d_in[4];
  const float* p11 = (const float*)d_in[5];
  const float* w12 = (const float*)d_in[6];
  const float* w15 = (const float*)d_in[7];
  const float* p16 = (const float*)d_in[8];
  float* out = (float*)d_out;

  const size_t SZ = (size_t)kN * kC * kHW;     // 6,422,528 floats
  float* ws  = (float*)d_ws;
  float* t3  = ws;
  float* t5  = t3 + SZ;
  float* t7  = t5 + SZ;
  float* t17 = t7 + SZ;
  float* t13 = t17 + SZ;
  float* t16 = t13 + SZ;
  float* t8  = t16 + SZ;
  float* t19 = t8 + (size_t)kN * kC * kC;
  float* t11 = t19 + (size_t)kN * kC * kC;

  const int total = (int)SZ;
  dim3 b256(256);
  dim3 wave(32);

  k_t3<<<(total + 255) / 256, b256, 0, stream>>>(x, p2, t3, total);
  k_softmax<<<(kN * kC * kW + 255) / 256, b256, 0, stream>>>(t3, t5);
  k_t7<<<dim3(kHW / 16, 1, kN), dim3(512), 0, stream>>>(x, w7, t7);
  k_t17<<<dim3(kHW / 16, kC / 16, kN), wave, 0, stream>>>(x, w15, t3, t17);
  k_t18<<<(total + 255) / 256, b256, 0, stream>>>(t7, t17, out, total);
  k_t8<<<dim3(kC / 16, kC / 16, kN), wave, 0, stream>>>(t5, t3, t8);
  k_t16<<<dim3(kHW / 16, kC / 16, kN), wave, 0, stream>>>(t8, p16, t16);
  k_t19<<<dim3(kC / 16, kC / 16, kN), wave, 0, stream>>>(t17, t16, t19);
  k_t11<<<(kN * kHW + 127) / 128, dim3(128), 0, stream>>>(t3, w6, p9, p11, t11);
  k_t13<<<(total + 255) / 256, b256, 0, stream>>>(t5, t7, w12, t11, t13, total);
  k_t20<<<dim3(kHW / 16, kC / 16, kN), wave, 0, stream>>>(t19, t13, out);
}